// MultiHeadAttention_515396076443
// MI455X (gfx1250) — compile-verified
//
#include <hip/hip_runtime.h>
#include <hip/hip_bf16.h>

// ---------------------------------------------------------------------------
// Problem constants (from reference)
// ---------------------------------------------------------------------------
#define B_    4
#define LQ_   1024
#define LK_   1024
#define H_    12
#define DK_   64
#define DV_   768
#define E_    768
#define HDV_  (H_ * DV_)   // 9216
#define HDK_  (H_ * DK_)   // 768
#define TOPK_ 64

typedef __attribute__((ext_vector_type(2))) float v2f;
typedef __attribute__((ext_vector_type(4))) float f4v;
typedef __attribute__((ext_vector_type(8))) float v8f;

// ---- CDNA5 async global->LDS staging (guarded: falls back if absent) ------
#if __has_builtin(__builtin_amdgcn_global_load_async_to_lds_b128) && \
    __has_builtin(__builtin_amdgcn_s_wait_asynccnt)
#define USE_ASYNC_LDS 1
// Builtin signature (from hipcc diagnostic): param0 = int4 AS(1)*, so pass
// typed vector pointers. AS1 value == flat value; AS3 offset == low 32 bits
// of the generic LDS pointer (ISA: LDS aperture maps addr[31:0] -> offset).
typedef int v4i __attribute__((vector_size(16)));
typedef __attribute__((address_space(1))) v4i gv4i_t;
typedef __attribute__((address_space(3))) v4i lv4i_t;
#define GLOBAL_AS(p) ((gv4i_t*)(unsigned long long)(p))
#define LDS_AS(p)    ((lv4i_t*)(unsigned int)(unsigned long long)(p))
#else
#define USE_ASYNC_LDS 0
#endif

// ---------------------------------------------------------------------------
// Generic WMMA-f32 GEMM:  C[M,N] = A[M,K] @ B[K,N]  (+ residual)
// If gidx != nullptr, A is a virtual gathered matrix (straight-through argmax
// attention made the forward attn one-hot, so attn@vh is a row gather):
//   A(m, c) = gsrc[(b*LK + gidx[(b*H + c/DV)*LQ + q]) * HDV + c],  m = b*LQ + q
// Block tile: 64(M) x 64(N), K panel 16. 8 waves in a 4(M) x 2(N) grid;
// each wave owns a 16x32 slab -> two v8f accumulators, WMMA f32 16x16x4.
// Panels staged into LDS via GLOBAL_LOAD_ASYNC_TO_LDS_B128 when available.
// ---------------------------------------------------------------------------
__global__ __launch_bounds__(256) void gemm_wmma_f32(
    const float* __restrict__ A, const float* __restrict__ Bm,
    float* __restrict__ C, int M, int N, int K,
    const float* __restrict__ residual,
    const int* __restrict__ gidx, const float* __restrict__ gsrc)
{
    __shared__ float As[64][20];  // 64 x 16 K-panel (+pad, 16B-aligned rows)
    __shared__ float Bs[16][68];  // 16 x 64 K-panel (+pad)

    const int tid  = threadIdx.x;
    const int m0   = blockIdx.y * 64;
    const int n0   = blockIdx.x * 64;
    const int w    = tid >> 5;          // wave 0..7
    const int lane = tid & 31;
    const int wm   = w >> 1;            // 0..3  (16-row slab)
    const int wn   = w & 1;             // 0..1  (32-col slab)
    const int lh   = lane & 15;         // lane % 16
    const int kha  = (lane >> 4) << 1;  // 0 or 2 (K half select for A/B frags)

    // tile-loader coordinates (one b128 per thread per panel)
    const int a_row = tid >> 2;         // 0..63
    const int a_col = (tid & 3) << 2;   // 0,4,8,12
    const int b_row = tid >> 4;         // 0..15
    const int b_col = (tid & 15) << 2;  // 0..60

    v8f acc0 = {}; v8f acc1 = {};

    for (int k0 = 0; k0 < K; k0 += 16) {
        // ---- stage A panel (optionally gathered through argmax indices) ----
        const float* asrc;
        if (gidx) {
            const int m = m0 + a_row;
            const int b = m / LQ_, q = m % LQ_;
            const int h = k0 / DV_;     // 768 % 16 == 0: panel stays in one head chunk
            const int kidx = gidx[(b * H_ + h) * LQ_ + q];
            asrc = gsrc + (size_t)(b * LK_ + kidx) * HDV_ + k0 + a_col;
        } else {
            asrc = A + (size_t)(m0 + a_row) * K + k0 + a_col;
            if (k0 + 16 < K) __builtin_prefetch(asrc + 16, 0, 0);  // global_prefetch_b8
        }
        const float* bsrc = Bm + (size_t)(k0 + b_row) * N + n0 + b_col;
        if (k0 + 16 < K) __builtin_prefetch(bsrc + (size_t)16 * N, 0, 0);

#if USE_ASYNC_LDS
        __builtin_amdgcn_global_load_async_to_lds_b128(
            GLOBAL_AS(asrc), LDS_AS(&As[a_row][a_col]), 0, 0);
        __builtin_amdgcn_global_load_async_to_lds_b128(
            GLOBAL_AS(bsrc), LDS_AS(&Bs[b_row][b_col]), 0, 0);
        __builtin_amdgcn_s_wait_asynccnt(0);
#else
        *reinterpret_cast<f4v*>(&As[a_row][a_col]) =
            *reinterpret_cast<const f4v*>(asrc);
        *reinterpret_cast<f4v*>(&Bs[b_row][b_col]) =
            *reinterpret_cast<const f4v*>(bsrc);
#endif
        __syncthreads();

        // ---- 4 WMMA K-substeps per panel ----
        #pragma unroll
        for (int ks = 0; ks < 16; ks += 4) {
            v2f a, b0, b1;
            a.x  = As[wm * 16 + lh][ks + kha];
            a.y  = As[wm * 16 + lh][ks + kha + 1];
            b0.x = Bs[ks + kha    ][wn * 32 + lh];
            b0.y = Bs[ks + kha + 1][wn * 32 + lh];
            b1.x = Bs[ks + kha    ][wn * 32 + 16 + lh];
            b1.y = Bs[ks + kha + 1][wn * 32 + 16 + lh];
            acc0 = __builtin_amdgcn_wmma_f32_16x16x4_f32(false, a, false, b0,
                                                         (short)0, acc0, false, false);
            acc1 = __builtin_amdgcn_wmma_f32_16x16x4_f32(false, a, false, b1,
                                                         (short)0, acc1, false, false);
        }
        __syncthreads();
    }

    // ---- epilogue: D layout = VGPR r, M = r + 8*(lane>=16), N = lane%16 ----
    const int mrow0 = m0 + wm * 16 + ((lane >> 4) << 3);
    #pragma unroll
    for (int r = 0; r < 8; ++r) {
        const int m  = mrow0 + r;
        const int na = n0 + wn * 32 + lh;
        const int nb = na + 16;
        float c0 = acc0[r], c1 = acc1[r];
        if (residual) {
            c0 += residual[(size_t)m * N + na];
            c1 += residual[(size_t)m * N + nb];
        }
        C[(size_t)m * N + na] = c0;
        C[(size_t)m * N + nb] = c1;
    }
}

// ---------------------------------------------------------------------------
// Scores + top-k-masked argmax.
// Grid: (LQ/16, H, B). Block 256 = 8 waves.
// Each wave WMMA-computes a 16x128 strip of S = (qh @ kh^T)/8 + qk_mask + k_mask
// into 64KB LDS, then owns 2 rows for selection: repeatedly extract the row
// max (first-index tie-break, matching jnp.argmax/top_k order); the first one
// with post_softmax_mask != 0 inside the top-64 wins (softmax is monotonic,
// renorm preserves order, so this IS the argmax of the reference's attn).
// ---------------------------------------------------------------------------
__global__ __launch_bounds__(256) void scores_topk_argmax(
    const float* __restrict__ qh, const float* __restrict__ kh,
    const float* __restrict__ qk_mask, const float* __restrict__ k_mask,
    const float* __restrict__ post_mask, int* __restrict__ idx_out)
{
    __shared__ float S[16][LK_];   // 65536 bytes

    const int q0   = blockIdx.x * 16;
    const int h    = blockIdx.y;
    const int b    = blockIdx.z;
    const int tid  = threadIdx.x;
    const int w    = tid >> 5;
    const int lane = tid & 31;
    const int lh   = lane & 15;
    const int kha  = (lane >> 4) << 1;

    // preload the 16 A fragments for this q-tile (A row = qh[b, q0+lh, h, :])
    v2f afrag[16];
    const float* qbase = qh + ((size_t)(b * LQ_ + q0 + lh) * H_ + h) * DK_;
    #pragma unroll
    for (int kk = 0; kk < 16; ++kk) {
        afrag[kk].x = qbase[kk * 4 + kha];
        afrag[kk].y = qbase[kk * 4 + kha + 1];
    }

    // 8 N-tiles of 16 columns per wave
    for (int t = 0; t < 8; ++t) {
        const int n0 = w * 128 + t * 16;
        const float* kbase = kh + ((size_t)(b * LK_ + n0 + lh) * H_ + h) * DK_;
        v8f acc = {};
        #pragma unroll
        for (int kk = 0; kk < 16; ++kk) {
            v2f bfrag;   // B[d][n] = kh[n][d]
            bfrag.x = kbase[kk * 4 + kha];
            bfrag.y = kbase[kk * 4 + kha + 1];
            acc = __builtin_amdgcn_wmma_f32_16x16x4_f32(false, afrag[kk], false, bfrag,
                                                        (short)0, acc, false, false);
        }
        // epilogue: 1/sqrt(DK) scale + additive masks -> LDS
        const int col   = n0 + lh;
        const float km  = k_mask[(size_t)b * LK_ + col];
        const int rbase = (lane >> 4) << 3;
        #pragma unroll
        for (int r = 0; r < 8; ++r) {
            const int qr = rbase + r;
            S[qr][col] = acc[r] * 0.125f
                       + qk_mask[((size_t)b * LQ_ + q0 + qr) * LK_ + col] + km;
        }
    }
    __syncthreads();

    // selection: wave w handles rows 2w and 2w+1
    for (int rr = 0; rr < 2; ++rr) {
        const int row = w * 2 + rr;
        const int q   = q0 + row;
        const float* pm = post_mask + ((size_t)b * LQ_ + q) * LK_;

        int result = -1;
        for (int it = 0; it < TOPK_ && result < 0; ++it) {
            float best = -3.0e38f; int bidx = LK_;
            for (int c = lane; c < LK_; c += 32) {
                const float v = S[row][c];
                if (v > best) { best = v; bidx = c; }
            }
            #pragma unroll
            for (int off = 16; off > 0; off >>= 1) {
                const float ov = __shfl_xor(best, off, 32);
                const int   oi = __shfl_xor(bidx, off, 32);
                if (ov > best || (ov == best && oi < bidx)) { best = ov; bidx = oi; }
            }
            if (pm[bidx] != 0.0f) result = bidx;           // uniform across the wave
            else if (lane == 0)   S[row][bidx] = -3.0e38f; // knock out, take next-largest
        }
        if (result < 0) {  // all top-64 masked: softmax tail ties -> first unmasked
            int first = LK_;
            for (int c = lane; c < LK_; c += 32)
                if (pm[c] != 0.0f) { first = c; break; }
            #pragma unroll
            for (int off = 16; off > 0; off >>= 1) {
                const int oi = __shfl_xor(first, off, 32);
                if (oi < first) first = oi;
            }
            result = (first < LK_) ? first : 0;
        }
        if (lane == 0) idx_out[((size_t)b * H_ + h) * LQ_ + q] = result;
    }
}

// ---------------------------------------------------------------------------
// One-hot attn writeout: single fused streaming pass with non-temporal stores
// (201 MB written once, never re-read -> NT keeps vh resident in the 192MB L2
// for the subsequent fc GEMM). idx[row] is wave-uniform and cache-resident.
// ---------------------------------------------------------------------------
__global__ __launch_bounds__(256) void write_onehot(
    const int* __restrict__ idx, f4v* __restrict__ attn4, long n4)
{
    long i = (long)blockIdx.x * blockDim.x + threadIdx.x;
    const long stride = (long)gridDim.x * blockDim.x;
    for (; i < n4; i += stride) {
        const long row = i >> 8;                 // LK_/4 = 256 float4 per row
        const int  c0  = (int)(i & 255) << 2;
        const int  d   = idx[row] - c0;          // position of the 1.0 in this chunk
        f4v val;
        val.x = (d == 0) ? 1.0f : 0.0f;
        val.y = (d == 1) ? 1.0f : 0.0f;
        val.z = (d == 2) ? 1.0f : 0.0f;
        val.w = (d == 3) ? 1.0f : 0.0f;
        __builtin_nontemporal_store(val, attn4 + i);
    }
}

// ---------------------------------------------------------------------------
// Launcher
// ---------------------------------------------------------------------------
extern "C" void kernel_launch(void* const* d_in, const int* in_sizes, int n_in,
                              void* d_out, int out_size, void* d_ws, size_t ws_size,
                              hipStream_t stream)
{
    (void)in_sizes; (void)n_in; (void)out_size; (void)ws_size;

    const float* q         = (const float*)d_in[0];
    const float* k         = (const float*)d_in[1];
    const float* v         = (const float*)d_in[2];
    /* d_in[3] qpos, d_in[4] kpos: unused by the math */
    const float* qk_mask   = (const float*)d_in[5];
    const float* k_mask    = (const float*)d_in[6];
    const float* post_mask = (const float*)d_in[7];
    const float* w_qs      = (const float*)d_in[8];
    const float* w_ks      = (const float*)d_in[9];
    const float* w_vs      = (const float*)d_in[10];
    const float* fc        = (const float*)d_in[11];

    float* out      = (float*)d_out;                       // [B, LQ, DV]
    float* attn_out = out + (size_t)B_ * LQ_ * DV_;        // [B, H, LQ, LK]

    // workspace: qh(12.6MB) + kh(12.6MB) + vh(151MB) + idx(192KB)
    char* ws = (char*)d_ws;
    float* qh = (float*)ws;  ws += (size_t)B_ * LQ_ * HDK_ * sizeof(float);
    float* kh = (float*)ws;  ws += (size_t)B_ * LK_ * HDK_ * sizeof(float);
    float* vh = (float*)ws;  ws += (size_t)B_ * LK_ * HDV_ * sizeof(float);
    int*   ix = (int*)ws;

    const dim3 blk(256);
    const int M = B_ * LQ_;   // 4096

    // projections
    gemm_wmma_f32<<<dim3(HDK_ / 64, M / 64), blk, 0, stream>>>(
        q, w_qs, qh, M, HDK_, E_, nullptr, nullptr, nullptr);
    gemm_wmma_f32<<<dim3(HDK_ / 64, M / 64), blk, 0, stream>>>(
        k, w_ks, kh, M, HDK_, E_, nullptr, nullptr, nullptr);
    gemm_wmma_f32<<<dim3(HDV_ / 64, M / 64), blk, 0, stream>>>(
        v, w_vs, vh, M, HDV_, DV_, nullptr, nullptr, nullptr);

    // scores + masked top-k argmax
    scores_topk_argmax<<<dim3(LQ_ / 16, H_, B_), blk, 0, stream>>>(
        qh, kh, qk_mask, k_mask, post_mask, ix);

    // hard one-hot attention output (fused zero+scatter, NT streaming stores)
    write_onehot<<<4096, 256, 0, stream>>>(
        ix, (f4v*)attn_out, (long)B_ * H_ * LQ_ * LK_ / 4);

    // output projection with fused gather of vh rows + residual v
    gemm_wmma_f32<<<dim3(DV_ / 64, M / 64), blk, 0, stream>>>(
        nullptr, fc, out, M, DV_, HDV_, v, ix, vh);
}